// skipgram_8675833938433
// MI455X (gfx1250) — compile-verified
//
#include <hip/hip_runtime.h>
#include <math.h>

// ---------------------------------------------------------------------------
// Skip-gram negative-sampling loss for MI455X (gfx1250, wave32).
// Memory-bound gather (≈100 MB random 512B rows @ 23.3 TB/s ≈ 4.3 us floor).
//  - Row gather: GLOBAL_LOAD_ASYNC_TO_LDS_B128 (direct HBM->LDS, ASYNCcnt)
//  - Dot products: V_WMMA_F32_16X16X4_F32 (exact fp32 matrix path)
// ---------------------------------------------------------------------------

typedef __attribute__((ext_vector_type(2))) float v2f;
typedef __attribute__((ext_vector_type(8))) float v8f;

#define DIM 128
#define ROW_STRIDE 132          // pad 128 -> 132 floats: 32 distinct LDS banks
#define ROWS_PER_WAVE 16
#define WAVES_PER_BLOCK 4
#define THREADS (WAVES_PER_BLOCK * 32)
#define B_SEG 65536
#define TILES_PER_SEG (B_SEG / ROWS_PER_WAVE)       // 4096
#define N_TILES (3 * TILES_PER_SEG)                 // 12288
#define N_BLOCKS (N_TILES / WAVES_PER_BLOCK)        // 3072

__global__ __launch_bounds__(THREADS)
void sg_score_kernel(const int* __restrict__ u_pos,
                     const int* __restrict__ v_pos,
                     const int* __restrict__ v_neg_city,
                     const int* __restrict__ v_neg_country,
                     const float* __restrict__ u_weight,
                     const float* __restrict__ v_weight,
                     float* __restrict__ partials)
{
    __shared__ float uS[DIM];
    __shared__ float rows[WAVES_PER_BLOCK][ROWS_PER_WAVE][ROW_STRIDE];
    __shared__ float wsum[WAVES_PER_BLOCK];

    const int tid  = threadIdx.x;
    const int lane = tid & 31;
    const int wave = tid >> 5;

    // Stage center embedding u (hot in L2 across all blocks).
    if (tid < DIM) {
        uS[tid] = u_weight[(size_t)u_pos[0] * DIM + tid];
    }
    __syncthreads();

    // One 16-row tile per wave; segments are multiples of 16 rows, so the
    // index array and softplus sign are wave-uniform.
    const int tile = blockIdx.x * WAVES_PER_BLOCK + wave;
    const int seg  = tile / TILES_PER_SEG;
    const int tIn  = tile - seg * TILES_PER_SEG;
    const int* idxArr = (seg == 0) ? v_pos : ((seg == 1) ? v_neg_city : v_neg_country);
    const int rowBase = tIn * ROWS_PER_WAVE;

    int myIdx = 0;
    if (lane < ROWS_PER_WAVE) myIdx = idxArr[rowBase + lane];

    // Gather 16 rows directly HBM -> LDS: per row one async B128 op, 32 lanes
    // x 16B = 512B fully coalesced, no VGPR round-trip. Tracked by ASYNCcnt.
    float (*myRows)[ROW_STRIDE] = rows[wave];
#pragma unroll
    for (int r = 0; r < ROWS_PER_WAVE; ++r) {
        const int ridx = __shfl(myIdx, r, 32);
        const unsigned long long gaddr =
            (unsigned long long)(uintptr_t)(v_weight + (size_t)ridx * DIM + lane * 4);
        // Low 32 bits of the flat shared pointer == LDS byte offset (aperture
        // mapping discards the upper bits).
        const unsigned ldsOff = (unsigned)(uintptr_t)(&myRows[r][lane * 4]);
        asm volatile("global_load_async_to_lds_b128 %0, %1, off"
                     :
                     : "v"(ldsOff), "v"(gaddr)
                     : "memory");
    }
    // Wait for all 16 async HBM->LDS transfers of this wave.
    asm volatile("s_wait_asynccnt 0" ::: "memory");

    // WMMA fragments (fp32 16x16x4):
    //   A (16x4): u[kb..kb+3] replicated over M; lane<16 holds K={0,1},
    //             lane>=16 holds K={2,3}  -> float2 at uS[kb + 2h].
    //   B (4x16): column j = gathered row j -> float2 at myRows[j][kb + 2h].
    //   D[i][j]  = sum_k u[k] * V_j[k]  (identical for all i).
    const int j = lane & 15;
    const int h = lane >> 4;

    v8f c = {0.f, 0.f, 0.f, 0.f, 0.f, 0.f, 0.f, 0.f};
#pragma unroll 8
    for (int kb = 0; kb < DIM; kb += 4) {
        v2f a = *(const v2f*)&uS[kb + 2 * h];
        v2f b = *(const v2f*)&myRows[j][kb + 2 * h];
        c = __builtin_amdgcn_wmma_f32_16x16x4_f32(
                /*neg_a=*/false, a, /*neg_b=*/false, b,
                /*c_mod=*/(short)0, c, /*reuse_a=*/false, /*reuse_b=*/false);
    }

    // Lanes 0..15 of c[0] hold the 16 scores (M=0 row of D).
    const float score = c[0];
    float sp = 0.f;
    if (lane < 16) {
        // loss contribution: softplus(-pos) for seg 0, softplus(+neg) else.
        // softplus(x) = max(x,0) + log(1 + exp(-|x|)); log arg in (1,2].
        const float x = (seg == 0) ? -score : score;
        sp = fmaxf(x, 0.f) + __logf(1.f + __expf(-fabsf(x)));
    }
    // Wave reduction (upper 16 lanes contribute 0).
#pragma unroll
    for (int off = 16; off >= 1; off >>= 1) sp += __shfl_xor(sp, off, 32);
    if (lane == 0) wsum[wave] = sp;
    __syncthreads();

    if (tid == 0) {
        float s = 0.f;
#pragma unroll
        for (int w = 0; w < WAVES_PER_BLOCK; ++w) s += wsum[w];
        partials[blockIdx.x] = s;
    }
}

// Deterministic fixed-order final reduction of block partials.
__global__ __launch_bounds__(256)
void sg_reduce_kernel(const float* __restrict__ partials, float* __restrict__ out)
{
    __shared__ float buf[256];
    float acc = 0.f;
    for (int i = threadIdx.x; i < N_BLOCKS; i += 256) acc += partials[i];
    buf[threadIdx.x] = acc;
    __syncthreads();
    for (int s = 128; s >= 1; s >>= 1) {
        if ((int)threadIdx.x < s) buf[threadIdx.x] += buf[threadIdx.x + s];
        __syncthreads();
    }
    if (threadIdx.x == 0) out[0] = buf[0];
}

extern "C" void kernel_launch(void* const* d_in, const int* in_sizes, int n_in,
                              void* d_out, int out_size, void* d_ws, size_t ws_size,
                              hipStream_t stream)
{
    const int*   u_pos         = (const int*)d_in[0];
    const int*   v_pos         = (const int*)d_in[1];
    const int*   v_neg_city    = (const int*)d_in[2];
    const int*   v_neg_country = (const int*)d_in[3];
    const float* u_weight      = (const float*)d_in[4];
    const float* v_weight      = (const float*)d_in[5];
    float*       out           = (float*)d_out;
    float*       partials      = (float*)d_ws;   // N_BLOCKS floats (12 KB)

    sg_score_kernel<<<N_BLOCKS, THREADS, 0, stream>>>(
        u_pos, v_pos, v_neg_city, v_neg_country, u_weight, v_weight, partials);
    sg_reduce_kernel<<<1, 256, 0, stream>>>(partials, out);
}